// GCN_17763984736424
// MI455X (gfx1250) — compile-verified
//
#include <hip/hip_runtime.h>

#define D_FEAT 64
#define TILE   256
#define EDGE_GRID 1024

// ---------------------------------------------------------------------------
// gfx1250 async global->LDS staging (CDNA5 ASYNCcnt path)
// ---------------------------------------------------------------------------
#if defined(__AMDGCN__) && __has_builtin(__builtin_amdgcn_global_load_async_to_lds_b32) && \
    __has_builtin(__builtin_amdgcn_s_wait_asynccnt)
#define USE_ASYNC_LDS 1
#if __has_builtin(__builtin_amdgcn_global_load_async_to_lds_b128)
#define USE_ASYNC_B128 1
#endif
#endif

// b32 builtin: (addrspace(1) int*, addrspace(3) int*, imm, imm)
typedef __attribute__((address_space(1))) int gint_t;
typedef __attribute__((address_space(3))) int lint_t;
// b128 builtin: (addrspace(1) int4*, addrspace(3) int4*, imm, imm)
typedef int v4i __attribute__((vector_size(16)));
typedef __attribute__((address_space(1))) v4i gv4i_t;
typedef __attribute__((address_space(3))) v4i lv4i_t;

__device__ static inline gint_t* as_global(const void* p) {
  return (gint_t*)(unsigned long long)p;
}
__device__ static inline lint_t* as_lds(void* p) {
  // generic shared address: low 32 bits are the workgroup-relative LDS offset
  return (lint_t*)(unsigned int)(unsigned long long)p;
}
__device__ static inline gv4i_t* as_global4(const void* p) {
  return (gv4i_t*)(unsigned long long)p;
}
__device__ static inline lv4i_t* as_lds4(void* p) {
  return (lv4i_t*)(unsigned int)(unsigned long long)p;
}

__device__ static inline int imin(int a, int b) { return a < b ? a : b; }

// ---------------------------------------------------------------------------
// threefry2x32 (JAX PRNG) -- folded key evaluated at compile time
// ---------------------------------------------------------------------------
struct KP { unsigned a, b; };

__host__ __device__ constexpr unsigned rotl32(unsigned x, int n) {
  return (x << n) | (x >> (32 - n));
}

__host__ __device__ constexpr KP threefry2x32(unsigned k0, unsigned k1,
                                              unsigned x0, unsigned x1) {
  unsigned ks[3] = {k0, k1, k0 ^ k1 ^ 0x1BD11BDAu};
  x0 += ks[0]; x1 += ks[1];
  const int rots[2][4] = {{13, 15, 26, 6}, {17, 29, 16, 24}};
  for (int i = 0; i < 5; ++i) {
    const int* r = rots[i & 1];
    for (int j = 0; j < 4; ++j) { x0 += x1; x1 = rotl32(x1, r[j]); x1 ^= x0; }
    x0 += ks[(i + 1) % 3];
    x1 += ks[(i + 2) % 3] + (unsigned)(i + 1);
  }
  return {x0, x1};
}

// key = fold_in(key(0), 1) = threefry2x32((0,0), (0,1))
constexpr KP FOLDED_KEY = threefry2x32(0u, 0u, 0u, 1u);

__device__ static inline void atomic_add_f32(float* p, float v) {
  (void)__hip_atomic_fetch_add(p, v, __ATOMIC_RELAXED, __HIP_MEMORY_SCOPE_AGENT);
}

// ---------------------------------------------------------------------------
// Elementwise / setup kernels
// ---------------------------------------------------------------------------
__global__ __launch_bounds__(256) void k_copy_w(const float* __restrict__ ew,
                                                float* __restrict__ w, int E) {
  int i = blockIdx.x * blockDim.x + threadIdx.x;
  if (i < E) w[i] = ew[i];
}

// drop_size is even: thread i covers counter pair (i, i + half)
__global__ __launch_bounds__(256) void k_dropout(float* __restrict__ w, int half,
                                                 unsigned span) {
  int i = blockIdx.x * blockDim.x + threadIdx.x;
  if (i >= half) return;
  KP r = threefry2x32(FOLDED_KEY.a, FOLDED_KEY.b, (unsigned)i, (unsigned)(i + half));
  unsigned i0 = (unsigned)((unsigned long long)r.a % span);
  unsigned i1 = (unsigned)((unsigned long long)r.b % span);
  w[i0] = 0.0f;   // idempotent scattered zero-writes: race-free in effect
  w[i1] = 0.0f;
}

__global__ __launch_bounds__(256) void k_zero_deg(float* __restrict__ dO,
                                                  float* __restrict__ dI, int N) {
  int i = blockIdx.x * blockDim.x + threadIdx.x;
  if (i < N) { dO[i] = 0.0f; dI[i] = 0.0f; }
}

__global__ __launch_bounds__(256) void k_degree(const int* __restrict__ src,
                                                const int* __restrict__ dst,
                                                float* __restrict__ dO,
                                                float* __restrict__ dI, int E) {
  int i = blockIdx.x * blockDim.x + threadIdx.x;
  if (i < E) {
    atomic_add_f32(&dO[src[i]], 1.0f);
    atomic_add_f32(&dI[dst[i]], 1.0f);
  }
}

__global__ __launch_bounds__(256) void k_norm(float* __restrict__ dO,
                                              float* __restrict__ dI, int N) {
  int i = blockIdx.x * blockDim.x + threadIdx.x;
  if (i < N) {
    float a = dO[i], b = dI[i];
    dO[i] = rsqrtf(a > 1.0f ? a : 1.0f);
    dI[i] = rsqrtf(b > 1.0f ? b : 1.0f);
  }
}

// hpre = feature * norm_out[row]; acc = 0
__global__ __launch_bounds__(256) void k_pre1(const float* __restrict__ feat,
                                              const float* __restrict__ nO,
                                              float* __restrict__ hpre,
                                              float* __restrict__ acc, int Nd) {
  int i = blockIdx.x * blockDim.x + threadIdx.x;
  if (i < Nd) {
    hpre[i] = feat[i] * nO[i >> 6];
    acc[i]  = 0.0f;
  }
}

// ---------------------------------------------------------------------------
// Edge scatter: acc[dst] += hpre[src] * w.
// Persistent blocks, double-buffered async->LDS staging of edge triples,
// 16 lanes per edge each owning a float4 (16B) slice of the 256B row.
// ---------------------------------------------------------------------------
__global__ __launch_bounds__(256) void k_edges(const int* __restrict__ src,
                                               const int* __restrict__ dst,
                                               const float* __restrict__ w,
                                               const float* __restrict__ hpre,
                                               float* __restrict__ acc,
                                               int E, int ntiles) {
  __shared__ alignas(16) int   s_src[2][TILE];
  __shared__ alignas(16) int   s_dst[2][TILE];
  __shared__ alignas(16) float s_w[2][TILE];

  const int tid = threadIdx.x;

  // Stage one 256-edge tile into LDS buffer `buf` (async DMA where possible).
  auto stage = [&](int buf, int tile) {
    const int base = tile * TILE;
    const int n    = imin(TILE, E - base);
#if USE_ASYNC_B128
    const int nq = n >> 2;   // quads per array (<= 64); tiles are 1KB-aligned
    if (tid < nq) {
      const int q = tid << 2;
      __builtin_amdgcn_global_load_async_to_lds_b128(
          as_global4(src + base + q), as_lds4(&s_src[buf][q]), 0, 0);
    } else if (tid < 2 * nq) {
      const int q = (tid - nq) << 2;
      __builtin_amdgcn_global_load_async_to_lds_b128(
          as_global4(dst + base + q), as_lds4(&s_dst[buf][q]), 0, 0);
    } else if (tid < 3 * nq) {
      const int q = (tid - 2 * nq) << 2;
      __builtin_amdgcn_global_load_async_to_lds_b128(
          as_global4(w + base + q), as_lds4(&s_w[buf][q]), 0, 0);
    }
    const int rem = n & 3;   // tail (none for E % 4 == 0); direct stores
    if (tid < rem) {
      const int idx = (n & ~3) + tid;
      s_src[buf][idx] = src[base + idx];
      s_dst[buf][idx] = dst[base + idx];
      s_w[buf][idx]   = w[base + idx];
    }
#elif USE_ASYNC_LDS
    if (tid < n) {
      __builtin_amdgcn_global_load_async_to_lds_b32(
          as_global(src + base + tid), as_lds(&s_src[buf][tid]), 0, 0);
      __builtin_amdgcn_global_load_async_to_lds_b32(
          as_global(dst + base + tid), as_lds(&s_dst[buf][tid]), 0, 0);
      __builtin_amdgcn_global_load_async_to_lds_b32(
          as_global(w + base + tid), as_lds(&s_w[buf][tid]), 0, 0);
    }
#else
    if (tid < n) {
      s_src[buf][tid] = src[base + tid];
      s_dst[buf][tid] = dst[base + tid];
      s_w[buf][tid]   = w[base + tid];
    }
#endif
  };

  auto compute = [&](int buf, int n) {
    const int lane = tid & 15;   // float4 slice within the 64-float row
    for (int e = tid >> 4; e < n; e += 16) {
      const float ww = s_w[buf][e];
      if (ww != 0.0f) {          // ~50% of edges are dropped -> skip traffic
        const int srow = s_src[buf][e] << 6;
        const int drow = s_dst[buf][e] << 6;
        const float4 hv =
            *reinterpret_cast<const float4*>(hpre + srow + (lane << 2));
        float* a = acc + drow + (lane << 2);
        atomic_add_f32(a + 0, hv.x * ww);
        atomic_add_f32(a + 1, hv.y * ww);
        atomic_add_f32(a + 2, hv.z * ww);
        atomic_add_f32(a + 3, hv.w * ww);
      }
    }
  };

  int tile = blockIdx.x;
  if (tile >= ntiles) return;
  const int stride = gridDim.x;

  int buf = 0;
  stage(buf, tile);
  for (; tile < ntiles; tile += stride) {
#if USE_ASYNC_LDS
    __builtin_amdgcn_s_wait_asynccnt(0);   // current buffer's DMA complete
#endif
    __syncthreads();                        // visible to all waves
    const int next = tile + stride;
    if (next < ntiles) stage(buf ^ 1, next);  // overlap next DMA with compute
    compute(buf, imin(TILE, E - tile * TILE));
    __syncthreads();                        // consumers done before reuse
    buf ^= 1;
  }
}

// h1 = acc*norm_in; out = feature + h1; hpre = h1*norm_out; acc = 0
__global__ __launch_bounds__(256) void k_mid(const float* __restrict__ feat,
                                             float* __restrict__ acc,
                                             const float* __restrict__ nO,
                                             const float* __restrict__ nI,
                                             float* __restrict__ hpre,
                                             float* __restrict__ out, int Nd) {
  int i = blockIdx.x * blockDim.x + threadIdx.x;
  if (i < Nd) {
    const int row = i >> 6;
    const float h1 = acc[i] * nI[row];
    out[i]  = feat[i] + h1;
    hpre[i] = h1 * nO[row];
    acc[i]  = 0.0f;
  }
}

// out = (out + acc*norm_in) / 3
__global__ __launch_bounds__(256) void k_final(const float* __restrict__ acc,
                                               const float* __restrict__ nI,
                                               float* __restrict__ out, int Nd) {
  int i = blockIdx.x * blockDim.x + threadIdx.x;
  if (i < Nd) {
    out[i] = (out[i] + acc[i] * nI[i >> 6]) * (1.0f / 3.0f);
  }
}

// ---------------------------------------------------------------------------
// Launch
// ---------------------------------------------------------------------------
extern "C" void kernel_launch(void* const* d_in, const int* in_sizes, int n_in,
                              void* d_out, int out_size, void* d_ws, size_t ws_size,
                              hipStream_t stream) {
  const float* feature = (const float*)d_in[0];
  const float* edge_w  = (const float*)d_in[1];
  const int*   src     = (const int*)d_in[2];
  const int*   dst     = (const int*)d_in[3];

  const int Nd = in_sizes[0];          // N * 64
  const int E  = in_sizes[1];
  const int N  = Nd / D_FEAT;
  float* out = (float*)d_out;

  // workspace layout (floats): w[E] | normO[N] | normI[N] | hpre[Nd] | acc[Nd]
  float* ws   = (float*)d_ws;
  float* w    = ws;
  float* nO   = w  + (size_t)E;
  float* nI   = nO + (size_t)N;
  float* hpre = nI + (size_t)N;        // byte offset (E+2N)*4 is 16B-aligned
  float* acc  = hpre + (size_t)Nd;

  const int drop = (int)(0.7 * (double)E);   // 875000
  const int half = drop / 2;

  const int TB = 256;
  const int ntiles = (E + TILE - 1) / TILE;
  const int gedge  = ntiles < EDGE_GRID ? ntiles : EDGE_GRID;
  const dim3 gE((E + TB - 1) / TB), gN((N + TB - 1) / TB),
             gNd((Nd + TB - 1) / TB), gH((half + TB - 1) / TB);

  k_copy_w <<<gE,  TB, 0, stream>>>(edge_w, w, E);
  k_dropout<<<gH,  TB, 0, stream>>>(w, half, (unsigned)E);
  k_zero_deg<<<gN, TB, 0, stream>>>(nO, nI, N);
  k_degree <<<gE,  TB, 0, stream>>>(src, dst, nO, nI, E);
  k_norm   <<<gN,  TB, 0, stream>>>(nO, nI, N);

  k_pre1   <<<gNd, TB, 0, stream>>>(feature, nO, hpre, acc, Nd);
  k_edges  <<<gedge, TB, 0, stream>>>(src, dst, w, hpre, acc, E, ntiles);
  k_mid    <<<gNd, TB, 0, stream>>>(feature, acc, nO, nI, hpre, out, Nd);
  k_edges  <<<gedge, TB, 0, stream>>>(src, dst, w, hpre, acc, E, ntiles);
  k_final  <<<gNd, TB, 0, stream>>>(acc, nI, out, Nd);
}